// ImplicitModelLoRA2_41601053229645
// MI455X (gfx1250) — compile-verified
//
#include <hip/hip_runtime.h>
#include <math.h>

typedef float v2f __attribute__((ext_vector_type(2)));
typedef float v4f __attribute__((ext_vector_type(4)));
typedef float v8f __attribute__((ext_vector_type(8)));

#define N_ 512
#define K_ 32
#define P_ 1024
#define Q_ 512
#define M_ 2048
#define TM 32        // X columns handled per workgroup
#define XSTC 516     // column stride (floats) for Xcm/Zcm: 516 % 64 == 4 -> conflict-free b64
#define YST 36       // column stride (floats) for Ycm: 36 ~ 4*(9c mod 16) permutation
#define ITERS 300

static __device__ __forceinline__ v8f wmma4(v2f a, v2f b, v8f c) {
  // D = A(16x4 f32) * B(4x16 f32) + C(16x16 f32)
  return __builtin_amdgcn_wmma_f32_16x16x4_f32(
      /*neg_a=*/false, a, /*neg_b=*/false, b,
      /*c_mod=*/(short)0, c, /*reuse_a=*/false, /*reuse_b=*/false);
}

// ---------------------------------------------------------------------------
// Kernel 1: projection scales.
// ws[0] = scale for L   (inf-norm of L   vs rho = sqrt(0.95-0.45))
// ws[1] = scale for R^T (inf-norm of R^T vs rho) = max column-abs-sum of R
// ws[2] = scale for Diag (max|Diag| vs 0.45)
// ---------------------------------------------------------------------------
__global__ __launch_bounds__(512) void deq_scales_kernel(
    const float* __restrict__ L, const float* __restrict__ R,
    const float* __restrict__ Dg, float* __restrict__ ws) {
  __shared__ float red[512];
  const int t = threadIdx.x;
  const float rho = 0.70710678118654752f;

  // --- max abs row sum of L [512 x 32] ---
  float s = 0.f;
  for (int k = 0; k < K_; ++k) s += fabsf(L[t * K_ + k]);
  red[t] = s;
  __syncthreads();
  for (int off = 256; off > 0; off >>= 1) {
    if (t < off) red[t] = fmaxf(red[t], red[t + off]);
    __syncthreads();
  }
  if (t == 0) {
    float nrm = red[0];
    ws[0] = (nrm > rho) ? (rho / nrm) : 1.0f;
  }
  __syncthreads();

  // --- max abs column sum of R [512 x 32] (== rows of R^T) ---
  s = 0.f;
  if (t < K_) {
    for (int n = 0; n < N_; ++n) s += fabsf(R[n * K_ + t]);
  }
  red[t] = s;
  __syncthreads();
  for (int off = 256; off > 0; off >>= 1) {
    if (t < off) red[t] = fmaxf(red[t], red[t + off]);
    __syncthreads();
  }
  if (t == 0) {
    float nrm = red[0];
    ws[1] = (nrm > rho) ? (rho / nrm) : 1.0f;
  }
  __syncthreads();

  // --- max |Diag| [512] ---
  red[t] = fabsf(Dg[t]);
  __syncthreads();
  for (int off = 256; off > 0; off >>= 1) {
    if (t < off) red[t] = fmaxf(red[t], red[t + off]);
    __syncthreads();
  }
  if (t == 0) {
    float nrm = red[0];
    ws[2] = (nrm > 0.45f) ? (0.45f / nrm) : 1.0f;
  }
}

// ---------------------------------------------------------------------------
// Kernel 2: persistent per-column-tile DEQ solve + fused output GEMM.
// One workgroup owns 32 columns of X [512 x 32], column-major in LDS.
//   Z_tile = B @ U^T(:, tile)                       (WMMA, K = 1024)
//   300x:  Y = Rtp @ X (rank-32 bottleneck, K=512)  (WMMA, split-K over waves)
//          X = relu(Lp @ Y + Dp.*X + Z)             (WMMA, K=32)
//   out(tile,:) = (C @ X + D @ U^T(:,tile))^T       (WMMA, K = 512 + 1024)
// A-fragments of Lp / Rtp are pre-swizzled [tile][kstep][lane] (v2f) ->
// coalesced conflict-free ds_load_b64. X/Z/Y are column-major with strides
// 516/516/36 so every B-fragment is a single conflict-free ds_load_b64 and
// D-fragment spills / elementwise updates are contiguous b128 DS ops.
// ---------------------------------------------------------------------------
__global__ __launch_bounds__(256) void deq_solve_kernel(
    const float* __restrict__ U,  const float* __restrict__ L,
    const float* __restrict__ R,  const float* __restrict__ Dg,
    const float* __restrict__ B,  const float* __restrict__ C,
    const float* __restrict__ D,  const float* __restrict__ ws,
    float* __restrict__ out) {
  extern __shared__ float smem[];
  float* Lsw = smem;                    // [32 mT][8 kk][32 lane] v2f = 16384 f
  float* Rsw = Lsw + N_ * K_;           // [2 yR][128 ks][32 lane] v2f = 16384 f
  float* Xcm = Rsw + K_ * N_;           // [32 col][516] state X     = 16512 f
  float* Zcm = Xcm + TM * XSTC;         // [32 col][516] Z tile      = 16512 f
  float* Ycm = Zcm + TM * XSTC;         // [2 kh][32 col][36]        =  2304 f
  float* Dps = Ycm + 2 * TM * YST;      // [512] clamped diag

  const int tid  = threadIdx.x;
  const int lane = tid & 31;
  const int wave = __builtin_amdgcn_readfirstlane(tid >> 5);  // scalar: 8 waves
  const int lr   = lane & 15;           // lane % 16
  const int lh   = lane >> 4;           // lane / 16 (0 or 1)
  const int m0   = blockIdx.x * TM;     // first global column of this tile

  const float sL = ws[0], sR = ws[1], sD = ws[2];

  // ---- stage swizzled A-fragments of Lp and Rtp into LDS ----
  // Rsw fragment i = (yR*128 + ks)*32 + lane:
  //   Rtp[yR*16 + lane%16][4ks + 2*(lane/16) + {0,1}] = R[n][krow]*sR
  for (int i = tid; i < 2 * 128 * 32; i += 256) {
    const int ln = i & 31;
    const int ks = (i >> 5) & 127;
    const int yR = i >> 12;
    const int krow = yR * 16 + (ln & 15);
    const int n = ks * 4 + 2 * (ln >> 4);
    v2f v;
    v.x = R[(size_t)n * K_ + krow] * sR;
    v.y = R[(size_t)(n + 1) * K_ + krow] * sR;
    ((v2f*)Rsw)[i] = v;
  }
  // Lsw fragment i = (mT*8 + kk)*32 + lane:
  //   Lp[mT*16 + lane%16][4kk + 2*(lane/16) + {0,1}]  (contiguous in global L)
  for (int i = tid; i < 32 * 8 * 32; i += 256) {
    const int ln = i & 31;
    const int kk = (i >> 5) & 7;
    const int mT = i >> 8;
    const int row = mT * 16 + (ln & 15);
    const int kp = kk * 4 + 2 * (ln >> 4);
    v2f v = *(const v2f*)(L + (size_t)row * K_ + kp);
    v.x *= sL;
    v.y *= sL;
    ((v2f*)Lsw)[i] = v;
  }
  for (int i = tid; i < N_; i += 256) Dps[i] = Dg[i] * sD;
  for (int i = tid; i < TM * XSTC; i += 256) Xcm[i] = 0.0f;  // X0 = 0

  // ---- Z[n][col] = sum_p B[n][p] * U[m0+col][p], stored column-major ----
  for (int t = wave * 8; t < wave * 8 + 8; ++t) {
    const int nT = t >> 1, jT = t & 1;
    const float* bRow = B + (size_t)(nT * 16 + lr) * P_ + 2 * lh;
    const float* uRow = U + (size_t)(m0 + jT * 16 + lr) * P_ + 2 * lh;
    v8f acc = {};
#pragma unroll 4
    for (int kp = 0; kp < P_ / 4; ++kp) {
      v2f a = *(const v2f*)(bRow + kp * 4);
      v2f b = *(const v2f*)(uRow + kp * 4);
      acc = wmma4(a, b, acc);
    }
    // lane holds rows nT*16 + 8*lh + (0..7) of column jT*16+lr: contiguous.
    float* zb = Zcm + (size_t)(jT * 16 + lr) * XSTC + nT * 16 + 8 * lh;
    v4f lo, hi;
    lo.x = acc[0]; lo.y = acc[1]; lo.z = acc[2]; lo.w = acc[3];
    hi.x = acc[4]; hi.y = acc[5]; hi.z = acc[6]; hi.w = acc[7];
    *(v4f*)zb = lo;
    *(v4f*)(zb + 4) = hi;
  }
  __syncthreads();

  // ---- Picard iteration: X = relu(Lp @ (Rtp @ X) + Dp.*X + Z) ----
  for (int it = 0; it < ITERS; ++it) {
    // (a) Y = Rtp @ X : [32x512]@[512x32]. 4 output tiles x 2 K-halves = 8 waves.
    {
      const int t  = wave >> 1;        // 0..3 (scalar)
      const int yR = t >> 1, yC = t & 1;
      const int kh = wave & 1;         // K half: n in [kh*256, kh*256+256)
      const v2f* aFrag = (const v2f*)Rsw + ((size_t)yR * 128 + kh * 64) * 32 + lane;
      const float* xcol = Xcm + (size_t)(yC * 16 + lr) * XSTC;
      v8f acc = {};
#pragma unroll 4
      for (int ks = 0; ks < 64; ++ks) {
        const int nb = kh * 256 + ks * 4 + 2 * lh;
        v2f a = aFrag[ks * 32];
        v2f b = *(const v2f*)(xcol + nb);   // single conflict-free b64
        acc = wmma4(a, b, acc);
      }
      float* yb = Ycm + (size_t)kh * (TM * YST) +
                  (size_t)(yC * 16 + lr) * YST + yR * 16 + 8 * lh;
      v4f lo, hi;
      lo.x = acc[0]; lo.y = acc[1]; lo.z = acc[2]; lo.w = acc[3];
      hi.x = acc[4]; hi.y = acc[5]; hi.z = acc[6]; hi.w = acc[7];
      *(v4f*)yb = lo;
      *(v4f*)(yb + 4) = hi;
    }
    __syncthreads();

    // (b) X = relu(Lp @ Y + Dp.*X + Z). 64 tiles / 8 waves; in-place is safe
    //     because each element is read+written only by its owning lane.
    for (int t = wave * 8; t < wave * 8 + 8; ++t) {
      const int mT = t >> 1, nT = t & 1;
      const v2f* aFrag = (const v2f*)Lsw + (size_t)mT * 8 * 32 + lane;
      const float* y0 = Ycm + (size_t)(nT * 16 + lr) * YST;
      const float* y1 = y0 + TM * YST;
      v8f acc = {};
#pragma unroll
      for (int kk = 0; kk < 8; ++kk) {
        const int kb = kk * 4 + 2 * lh;
        v2f a = aFrag[kk * 32];
        v2f b = *(const v2f*)(y0 + kb) + *(const v2f*)(y1 + kb);
        acc = wmma4(a, b, acc);
      }
      // elementwise: rows mT*16 + 8*lh + (0..7) of column nT*16+lr: contiguous.
      const int rbase = mT * 16 + 8 * lh;
      float* xb = Xcm + (size_t)(nT * 16 + lr) * XSTC + rbase;
      const float* zb = Zcm + (size_t)(nT * 16 + lr) * XSTC + rbase;
      v4f xv0 = *(const v4f*)xb,        xv1 = *(const v4f*)(xb + 4);
      v4f zv0 = *(const v4f*)zb,        zv1 = *(const v4f*)(zb + 4);
      v4f dv0 = *(const v4f*)(Dps + rbase), dv1 = *(const v4f*)(Dps + rbase + 4);
      v4f o0, o1;
#pragma unroll
      for (int v = 0; v < 4; ++v) {
        float a0 = acc[v]     + dv0[v] * xv0[v] + zv0[v];
        float a1 = acc[v + 4] + dv1[v] * xv1[v] + zv1[v];
        o0[v] = a0 > 0.0f ? a0 : 0.0f;
        o1[v] = a1 > 0.0f ? a1 : 0.0f;
      }
      *(v4f*)xb = o0;
      *(v4f*)(xb + 4) = o1;
    }
    __syncthreads();
  }

  // ---- out[m0+j][q] = sum_n C[q][n] X[n][j] + sum_p D[q][p] U[m0+j][p] ----
  for (int t = wave * 8; t < wave * 8 + 8; ++t) {
    const int qT = t >> 1, mT = t & 1;
    v8f acc = {};
    const float* cRow = C + (size_t)(qT * 16 + lr) * N_ + 2 * lh;
    const float* xcol = Xcm + (size_t)(mT * 16 + lr) * XSTC;
#pragma unroll 4
    for (int ks = 0; ks < N_ / 4; ++ks) {
      const int nb = ks * 4 + 2 * lh;
      v2f a = *(const v2f*)(cRow + ks * 4);
      v2f b = *(const v2f*)(xcol + nb);
      acc = wmma4(a, b, acc);
    }
    const float* dRow = D + (size_t)(qT * 16 + lr) * P_ + 2 * lh;
    const float* uRow = U + (size_t)(m0 + mT * 16 + lr) * P_ + 2 * lh;
#pragma unroll 4
    for (int ks = 0; ks < P_ / 4; ++ks) {
      v2f a = *(const v2f*)(dRow + ks * 4);
      v2f b = *(const v2f*)(uRow + ks * 4);
      acc = wmma4(a, b, acc);
    }
#pragma unroll
    for (int v = 0; v < 8; ++v) {
      const int q  = qT * 16 + v + 8 * lh;
      const int ml = mT * 16 + lr;
      out[(size_t)(m0 + ml) * Q_ + q] = acc[v];
    }
  }
}

extern "C" void kernel_launch(void* const* d_in, const int* in_sizes, int n_in,
                              void* d_out, int out_size, void* d_ws, size_t ws_size,
                              hipStream_t stream) {
  const float* U  = (const float*)d_in[0];  // [2048,1024]
  const float* L  = (const float*)d_in[1];  // [512,32]
  const float* R  = (const float*)d_in[2];  // [512,32]
  const float* Dg = (const float*)d_in[3];  // [512]
  const float* B  = (const float*)d_in[4];  // [512,1024]
  const float* C  = (const float*)d_in[5];  // [512,512]
  const float* D  = (const float*)d_in[6];  // [512,1024]
  float* ws  = (float*)d_ws;
  float* out = (float*)d_out;               // [2048,512] f32

  deq_scales_kernel<<<1, 512, 0, stream>>>(L, R, Dg, ws);

  const size_t lds_bytes =
      (size_t)(N_ * K_ + K_ * N_ + TM * XSTC + TM * XSTC + 2 * TM * YST + N_) *
      sizeof(float);  // 274432 B <= 320 KB
  deq_solve_kernel<<<M_ / TM, 256, lds_bytes, stream>>>(U, L, R, Dg, B, C, D,
                                                        ws, out);
}